// RGCNLayer_76424648065359
// MI455X (gfx1250) — compile-verified
//
#include <hip/hip_runtime.h>

#define IN_FEAT   128
#define OUT_FEAT  128
#define NB        16
#define SI        8
#define SO        8

typedef __attribute__((ext_vector_type(2))) float v2f;
typedef __attribute__((ext_vector_type(8))) float v8f;

// ---------------------------------------------------------------------------
// Kernel 1: zero the aggregation buffer (d_out is poisoned by the harness)
// ---------------------------------------------------------------------------
__global__ void rgcn_zero_kernel(float* __restrict__ p, long long n) {
    long long i = (long long)blockIdx.x * blockDim.x + threadIdx.x;
    if (i < n) p[i] = 0.0f;
}

// ---------------------------------------------------------------------------
// Kernel 2: per-edge block-diagonal message + atomic scatter to dst
// One wave (32 lanes) per edge; each lane produces 4 contiguous output feats.
// Features o0..o0+3 lie inside one 8-wide basis block (o0 % 8 in {0,4}).
// Weight row (4 KB) and h (25.6 MB) are L2-resident on MI455X (192 MB L2).
// ---------------------------------------------------------------------------
__global__ void rgcn_edge_kernel(const float* __restrict__ h,
                                 const float* __restrict__ weight,
                                 const int*   __restrict__ src,
                                 const int*   __restrict__ dst,
                                 const int*   __restrict__ etype,
                                 float*       __restrict__ agg,
                                 int E_) {
    long long tid = (long long)blockIdx.x * blockDim.x + threadIdx.x;
    int e = (int)(tid >> 5);
    if (e >= E_) return;
    int lane = (int)(tid & 31);

    int s = src[e];
    int d = dst[e];
    int r = etype[e];

    const float* x = h + (long long)s * IN_FEAT;
    const float* w = weight + (long long)r * (NB * SI * SO);

    int o0 = lane * 4;        // first of 4 output features handled by this lane
    int b  = o0 >> 3;         // basis index (all 4 feats share it)
    int oo = o0 & 7;          // column offset inside the 8x8 block (0 or 4)

    const float* xb = x + b * SI;           // 8 src features for this basis
    const float* wb = w + b * (SI * SO);    // 8x8 block, row-major [i][o]

    // load the 8 x-values as two float4s (16B aligned)
    float4 x0 = *(const float4*)(xb);
    float4 x1 = *(const float4*)(xb + 4);
    float xv[8] = {x0.x, x0.y, x0.z, x0.w, x1.x, x1.y, x1.z, x1.w};

    float acc0 = 0.f, acc1 = 0.f, acc2 = 0.f, acc3 = 0.f;
#pragma unroll
    for (int i = 0; i < SI; ++i) {
        float4 wr = *(const float4*)(wb + i * SO + oo);  // 16B aligned
        acc0 = fmaf(xv[i], wr.x, acc0);
        acc1 = fmaf(xv[i], wr.y, acc1);
        acc2 = fmaf(xv[i], wr.z, acc2);
        acc3 = fmaf(xv[i], wr.w, acc3);
    }

    float* outp = agg + (long long)d * OUT_FEAT + o0;
    atomicAdd(outp + 0, acc0);
    atomicAdd(outp + 1, acc1);
    atomicAdd(outp + 2, acc2);
    atomicAdd(outp + 3, acc3);
}

// ---------------------------------------------------------------------------
// Kernel 3: fused self-loop GEMM (h @ loop_weight) via V_WMMA_F32_16X16X4_F32
// + epilogue relu(agg * norm + loop_msg), in place over d_out.
// One wave computes one 16x16 output tile; K=128 in steps of 4 -> 32 WMMAs.
// VGPR layouts per CDNA5 ISA 7.12.2:
//   A (16x4 f32, 2 VGPRs): lanes 0-15 hold K=k,k+1 of row M=lane;
//                          lanes 16-31 hold K=k+2,k+3 of row M=lane-16.
//   B (4x16 f32, 2 VGPRs): lanes 0-15 hold rows K=k,k+1 at col N=lane;
//                          lanes 16-31 hold rows K=k+2,k+3 at col N=lane-16.
//   C/D (16x16 f32, 8 VGPRs): vgpr v, lanes 0-15 -> M=v; lanes 16-31 -> M=v+8.
// ---------------------------------------------------------------------------
__global__ void rgcn_final_kernel(const float* __restrict__ h,
                                  const float* __restrict__ lw,
                                  const float* __restrict__ norm,
                                  float*       __restrict__ out,  // in: agg, out: final
                                  int N_) {
    int wave  = (int)(((long long)blockIdx.x * blockDim.x + threadIdx.x) >> 5);
    int lane  = threadIdx.x & 31;
    int mTile = wave >> 3;       // 8 N-tiles per row of tiles (128/16)
    int nTile = wave & 7;
    int row0  = mTile * 16;
    if (row0 >= N_) return;      // wave-uniform: EXEC stays all-1s for WMMA

    int col0 = nTile * 16;
    int half = lane >> 4;        // 0: lanes 0-15, 1: lanes 16-31
    int l16  = lane & 15;

    // A-row for this lane (clamp for a possible ragged last tile; duplicated
    // rows are computed but masked at the store).
    int arow_i = row0 + l16;
    if (arow_i >= N_) arow_i = N_ - 1;
    const float* arow = h + (long long)arow_i * IN_FEAT;

    v8f c = {};
    for (int k = 0; k < IN_FEAT; k += 4) {
        v2f a, b;
        // A: two consecutive K values for this lane's half
        float2 av = *(const float2*)(arow + k + half * 2);
        a.x = av.x; a.y = av.y;
        // B: rows k+2*half and k+1+2*half at column col0+l16 (stride 128)
        const float* bp = lw + (long long)(k + half * 2) * OUT_FEAT + col0 + l16;
        b.x = bp[0];
        b.y = bp[OUT_FEAT];
        c = __builtin_amdgcn_wmma_f32_16x16x4_f32(
                /*neg_a=*/false, a, /*neg_b=*/false, b,
                /*c_mod=*/(short)0, c, /*reuse_a=*/false, /*reuse_b=*/false);
    }

    // Epilogue: out = relu(agg * norm + loop_msg). Divergence is fine here
    // (all WMMAs already issued).
    int col = col0 + l16;
#pragma unroll
    for (int v = 0; v < 8; ++v) {
        int row = row0 + v + half * 8;
        if (row < N_) {
            float nrm = norm[row];
            long long idx = (long long)row * OUT_FEAT + col;
            float val = fmaf(out[idx], nrm, c[v]);
            out[idx] = val > 0.0f ? val : 0.0f;
        }
    }
}

// ---------------------------------------------------------------------------
// Launch
// Inputs (setup_inputs order): h[N*128], norm[N], weight[R*1024],
// loop_weight[128*128], src[E], dst[E], etype[E]. d_out: N*128 f32.
// ---------------------------------------------------------------------------
extern "C" void kernel_launch(void* const* d_in, const int* in_sizes, int n_in,
                              void* d_out, int out_size, void* d_ws, size_t ws_size,
                              hipStream_t stream) {
    const float* h      = (const float*)d_in[0];
    const float* norm   = (const float*)d_in[1];
    const float* weight = (const float*)d_in[2];
    const float* lw     = (const float*)d_in[3];
    const int*   src    = (const int*)d_in[4];
    const int*   dst    = (const int*)d_in[5];
    const int*   etype  = (const int*)d_in[6];
    float* out = (float*)d_out;

    const int N_ = in_sizes[1];         // norm has N elements
    const int E_ = in_sizes[4];         // src has E elements
    const long long totOut = (long long)N_ * OUT_FEAT;

    // 1) zero the accumulator (d_out)
    {
        int threads = 256;
        int blocks  = (int)((totOut + threads - 1) / threads);
        rgcn_zero_kernel<<<blocks, threads, 0, stream>>>(out, totOut);
    }

    // 2) edge messages + atomic scatter (1 wave per edge)
    {
        long long totalThreads = (long long)E_ * 32;
        int threads = 256;
        int blocks  = (int)((totalThreads + threads - 1) / threads);
        rgcn_edge_kernel<<<blocks, threads, 0, stream>>>(h, weight, src, dst,
                                                         etype, out, E_);
    }

    // 3) fused WMMA self-loop GEMM + norm/add/relu epilogue
    {
        int mTiles = (N_ + 15) / 16;
        long long waves = (long long)mTiles * (OUT_FEAT / 16);
        int threads = 256;                       // 8 waves per block
        int blocks  = (int)((waves * 32 + threads - 1) / threads);
        rgcn_final_kernel<<<blocks, threads, 0, stream>>>(h, lw, norm, out, N_);
    }
}